// GraphConv_19593640805096
// MI455X (gfx1250) — compile-verified
//
#include <hip/hip_runtime.h>

typedef __attribute__((ext_vector_type(2))) float v2f;
typedef __attribute__((ext_vector_type(8))) float v8f;

#define D_FEAT 128

// ---------------------------------------------------------------- zeroing ----
__global__ void zero_f4_kernel(float4* __restrict__ p, long n4) {
    long i = (long)blockIdx.x * blockDim.x + threadIdx.x;
    if (i < n4) p[i] = make_float4(0.f, 0.f, 0.f, 0.f);
}

__global__ void zero_f1_kernel(float* __restrict__ p, int n) {
    int i = blockIdx.x * blockDim.x + threadIdx.x;
    if (i < n) p[i] = 0.f;
}

// ---------------------------------------------------------------- degree ----
__global__ void degree_kernel(const int* __restrict__ dst,
                              float* __restrict__ deg, int E) {
    int i = blockIdx.x * blockDim.x + threadIdx.x;
    if (i < E) unsafeAtomicAdd(&deg[dst[i]], 1.0f);
}

// norm = rsqrt(max(deg,1)) in place
__global__ void norm_kernel(float* __restrict__ deg, int n) {
    int i = blockIdx.x * blockDim.x + threadIdx.x;
    if (i < n) {
        float d = deg[i];
        d = d < 1.0f ? 1.0f : d;
        deg[i] = rsqrtf(d);
    }
}

// ---------------------------------------------------------------- scatter ----
// One wave (32 lanes) per edge; each lane handles 4 consecutive floats.
// agg[dst] += feat[src] * norm[src]  via hardware global_atomic_add_f32.
__global__ void scatter_kernel(const float* __restrict__ feat,
                               const int* __restrict__ src,
                               const int* __restrict__ dst,
                               const float* __restrict__ norm,
                               float* __restrict__ agg, int E) {
    long gid = (long)blockIdx.x * blockDim.x + threadIdx.x;
    int e = (int)(gid >> 5);
    int lane = (int)(gid & 31);
    if (e >= E) return;
    int s = src[e];
    int d = dst[e];
    float ns = norm[s];
    const float4 f = *(const float4*)(feat + (long)s * D_FEAT + lane * 4);
    float* out = agg + (long)d * D_FEAT + lane * 4;
    unsafeAtomicAdd(out + 0, f.x * ns);
    unsafeAtomicAdd(out + 1, f.y * ns);
    unsafeAtomicAdd(out + 2, f.z * ns);
    unsafeAtomicAdd(out + 3, f.w * ns);
}

// ------------------------------------------------------------------ GEMM ----
// In-place: reads agg rows (A tile fully into registers first), writes rst to
// the same rows. Weight (128x128 f32 = 64KB) staged in LDS once per block.
// Block = 256 threads = 8 waves; each wave owns 16 rows; grid covers N rows.
__global__ void gemm_wmma_kernel(float* __restrict__ agg_out,      // [N,128] in/out
                                 const float* __restrict__ weight, // [128,128]
                                 const float* __restrict__ bias,   // [128]
                                 const float* __restrict__ norm,   // [N]
                                 int n_nodes) {
    __shared__ float w_lds[D_FEAT * D_FEAT];  // 64 KB

    const int tid = threadIdx.x;
    // cooperative load of the weight matrix into LDS (float4, coalesced)
    for (int i = tid * 4; i < D_FEAT * D_FEAT; i += 256 * 4) {
        *(float4*)&w_lds[i] = *(const float4*)&weight[i];
    }
    __syncthreads();

    const int wave = tid >> 5;        // 0..7
    const int lane = tid & 31;
    const int half = lane >> 4;       // 0 | 1
    const int l16  = lane & 15;

    const int rowBase = blockIdx.x * 128 + wave * 16;

    // ---- load full A tile (16 rows x 128 K) into registers ----
    // ISA 16x4 f32 A layout: lanes 0-15 hold M=lane, K = k0+{0,1};
    //                        lanes 16-31 hold M=lane-16, K = k0+{2,3}.
    int arow = rowBase + l16;
    int arow_c = arow < n_nodes ? arow : (n_nodes - 1);  // clamp; dup rows never stored
    const float* aptr = agg_out + (long)arow_c * D_FEAT + half * 2;

    v2f A[32];
#pragma unroll
    for (int k = 0; k < 32; ++k) {
        A[k] = *(const v2f*)(aptr + k * 4);
    }

    v8f acc[8] = {};

    // ---- 8 N-tiles x 32 K-steps of v_wmma_f32_16x16x4_f32 ----
#pragma unroll
    for (int k = 0; k < 32; ++k) {
        const int k0 = k * 4 + half * 2;
#pragma unroll
        for (int n = 0; n < 8; ++n) {
            const int col = n * 16 + l16;
            v2f B;
            B.x = w_lds[(k0 + 0) * D_FEAT + col];
            B.y = w_lds[(k0 + 1) * D_FEAT + col];
            acc[n] = __builtin_amdgcn_wmma_f32_16x16x4_f32(
                /*neg_a=*/false, A[k], /*neg_b=*/false, B,
                /*c_mod=*/(short)0, acc[n], /*reuse_a=*/false, /*reuse_b=*/false);
        }
    }

    // ---- epilogue: rst = acc * norm[row] + bias[col] ----
    // C layout: VGPR j, lanes 0-15 -> M=j ; lanes 16-31 -> M=8+j.
    float nrm[8];
#pragma unroll
    for (int j = 0; j < 8; ++j) {
        int r = rowBase + j + half * 8;
        nrm[j] = (r < n_nodes) ? norm[r] : 0.f;
    }
#pragma unroll
    for (int n = 0; n < 8; ++n) {
        const int col = n * 16 + l16;
        const float b = bias[col];
#pragma unroll
        for (int j = 0; j < 8; ++j) {
            int r = rowBase + j + half * 8;
            if (r < n_nodes) {
                agg_out[(long)r * D_FEAT + col] = acc[n][j] * nrm[j] + b;
            }
        }
    }
}

// ---------------------------------------------------------------- launch ----
extern "C" void kernel_launch(void* const* d_in, const int* in_sizes, int n_in,
                              void* d_out, int out_size, void* d_ws, size_t ws_size,
                              hipStream_t stream) {
    const float* feat   = (const float*)d_in[0];
    const float* weight = (const float*)d_in[1];
    const float* bias   = (const float*)d_in[2];
    const int*   src    = (const int*)d_in[3];
    const int*   dst    = (const int*)d_in[4];
    float* out  = (float*)d_out;   // doubles as agg accumulator
    float* norm = (float*)d_ws;    // N floats of scratch (deg -> norm in place)

    const int N = in_sizes[0] / D_FEAT;
    const int E = in_sizes[3];

    // 1) zero agg accumulator (d_out) and deg (d_ws)
    long n4 = (long)N * D_FEAT / 4;
    zero_f4_kernel<<<(int)((n4 + 255) / 256), 256, 0, stream>>>((float4*)out, n4);
    zero_f1_kernel<<<(N + 255) / 256, 256, 0, stream>>>(norm, N);

    // 2) in-degree
    degree_kernel<<<(E + 255) / 256, 256, 0, stream>>>(dst, norm, E);

    // 3) norm = rsqrt(max(deg,1))
    norm_kernel<<<(N + 255) / 256, 256, 0, stream>>>(norm, N);

    // 4) scatter-sum: agg[dst] += feat[src]*norm[src]   (1 wave / edge)
    long threads = (long)E * 32;
    scatter_kernel<<<(int)((threads + 255) / 256), 256, 0, stream>>>(
        feat, src, dst, norm, out, E);

    // 5) in-place WMMA GEMM + epilogue
    gemm_wmma_kernel<<<(N + 127) / 128, 256, 0, stream>>>(out, weight, bias, norm, N);
}